// CollisionRegularizer_77025943486733
// MI455X (gfx1250) — compile-verified
//
#include <hip/hip_runtime.h>

typedef __attribute__((ext_vector_type(2))) float v2f;
typedef __attribute__((ext_vector_type(8))) float v8f;

#define TILE 64
#define EPSF 1e-8f

// ---------------------------------------------------------------------------
// Kernel 1: quaternion -> rotation matrix (exact reference formula, NO
// normalization), stored with stride 12 floats (16B-aligned rows for b128).
// ---------------------------------------------------------------------------
__global__ void quat2rot_kernel(const float* __restrict__ rot,
                                float* __restrict__ R, int P) {
    int p = blockIdx.x * blockDim.x + threadIdx.x;
    if (p >= P) return;
    float qw = rot[p * 4 + 0], qx = rot[p * 4 + 1];
    float qy = rot[p * 4 + 2], qz = rot[p * 4 + 3];
    float* r = R + (size_t)p * 12;
    r[0]  = 1.f - 2.f * qy * qy - 2.f * qz * qz;
    r[1]  = 2.f * qx * qy - 2.f * qz * qw;
    r[2]  = 2.f * qx * qz + 2.f * qy * qw;
    r[3]  = 2.f * qx * qy + 2.f * qz * qw;
    r[4]  = 1.f - 2.f * qx * qx - 2.f * qz * qz;
    r[5]  = 2.f * qy * qz - 2.f * qx * qw;
    r[6]  = 2.f * qx * qz - 2.f * qy * qw;
    r[7]  = 2.f * qy * qz + 2.f * qx * qw;
    r[8]  = 1.f - 2.f * qx * qx - 2.f * qy * qy;
    r[9]  = 0.f; r[10] = 0.f; r[11] = 0.f;
}

// ---------------------------------------------------------------------------
// Kernel 2: 64x64 pair tile per 256-thread block (8 wave32).
// dist^2 for each 16x16 sub-tile via V_WMMA_F32_16X16X4_F32:
//   A[n][k] = (x_n, y_n, z_n, 1)                 (16x4)
//   B[k][m] = (-2x_m, -2y_m, -2z_m, |x_m|^2)^T   (4x16)
//   C[n][m] = |x_n|^2
//   D = A*B + C = |x_n|^2 - 2 x_n.x_m + |x_m|^2 = dist^2
// WMMA operands are read branch-free from LDS arrays pre-packed in the exact
// per-lane-half layout (index h*64 + point), avoiding exec-mask divergence.
// ---------------------------------------------------------------------------
__global__ void __launch_bounds__(256)
tile_kernel(const float* __restrict__ xyz, const float* __restrict__ scales,
            const float* __restrict__ Rg, float* __restrict__ partials, int N) {
    // row side
    __shared__ float4 posr[TILE];        // (x, y, z, |x|^2)
    __shared__ float4 scr[TILE];         // (sx, sy, sz, 0)
    __shared__ float  n2r[TILE];         // |x|^2 (contiguous for C build)
    __shared__ float  Rr[TILE * 12];
    __shared__ float2 apack[2 * TILE];   // A operand: h=0 -> (x,y), h=1 -> (z,1)
    // column side
    __shared__ float4 posc[TILE];
    __shared__ float4 scc[TILE];
    __shared__ float  Rc[TILE * 12];
    __shared__ float2 bpack[2 * TILE];   // B operand: h=0 -> (-2x,-2y), h=1 -> (-2z,|x|^2)
    __shared__ float  red[256];

    const int tid     = threadIdx.x;
    const int rowBase = blockIdx.x * TILE;
    const int colBase = blockIdx.y * TILE;
    const long long pb = (long long)blockIdx.z * N;

    // ---- stage tile data into LDS (wave-uniform branches: 64 = 2 waves) ----
    if (tid < 64) {
        int p = tid;
        long long g = pb + rowBase + p;
        float x = xyz[g * 3 + 0], y = xyz[g * 3 + 1], z = xyz[g * 3 + 2];
        float n2 = x * x + y * y + z * z;
        posr[p] = make_float4(x, y, z, n2);
        n2r[p]  = n2;
        scr[p]  = make_float4(scales[g * 3 + 0], scales[g * 3 + 1],
                              scales[g * 3 + 2], 0.f);
        apack[p]        = make_float2(x, y);
        apack[64 + p]   = make_float2(z, 1.0f);
    } else if (tid < 128) {
        int p = tid - 64;
        long long g = pb + rowBase + p;
#pragma unroll
        for (int k = 0; k < 12; ++k) Rr[p * 12 + k] = Rg[g * 12 + k];
    } else if (tid < 192) {
        int p = tid - 128;
        long long g = pb + colBase + p;
        float x = xyz[g * 3 + 0], y = xyz[g * 3 + 1], z = xyz[g * 3 + 2];
        float n2 = x * x + y * y + z * z;
        posc[p] = make_float4(x, y, z, n2);
        scc[p]  = make_float4(scales[g * 3 + 0], scales[g * 3 + 1],
                              scales[g * 3 + 2], 0.f);
        bpack[p]        = make_float2(-2.f * x, -2.f * y);
        bpack[64 + p]   = make_float2(-2.f * z, n2);
    } else {
        int p = tid - 192;
        long long g = pb + colBase + p;
#pragma unroll
        for (int k = 0; k < 12; ++k) Rc[p * 12 + k] = Rg[g * 12 + k];
    }
    __syncthreads();

    const int wave = tid >> 5;
    const int lane = tid & 31;
    const int h    = lane >> 4;   // lane half (K-block / M-block selector)
    const int j    = lane & 15;   // M for A-operand, N (column m) for B/C/D
    float acc = 0.f;

#pragma unroll
    for (int t = 0; t < 2; ++t) {
        const int st = wave + t * 8;   // 16 sub-tiles, 8 waves -> 2 each
        const int sn = st >> 2;        // row sub-tile
        const int sm = st & 3;         // col sub-tile
        const int pA   = sn * 16 + j;  // A-matrix row point
        const int mloc = sm * 16 + j;  // this lane's column point

        // branch-free operand fetch (single ds_load_b64 each)
        float2 af = apack[h * 64 + pA];
        float2 bf = bpack[h * 64 + mloc];
        v2f a;  a.x  = af.x; a.y  = af.y;
        v2f bm; bm.x = bf.x; bm.y = bf.y;
        // C: VGPR r holds row M = r + 8*h  ->  |x_n|^2 (contiguous -> b128s)
        v8f cc;
#pragma unroll
        for (int r = 0; r < 8; ++r) cc[r] = n2r[sn * 16 + 8 * h + r];

        v8f d2 = __builtin_amdgcn_wmma_f32_16x16x4_f32(
            false, a, false, bm, (short)0, cc, false, false);

        // column-side (m) data is fixed for this lane: hoist out of r-loop
        const float4 pm  = posc[mloc];
        const float4 smv = scc[mloc];
        const float* Rm = &Rc[mloc * 12];
        const float Rm0 = Rm[0], Rm1 = Rm[1], Rm2 = Rm[2];
        const float Rm3 = Rm[3], Rm4 = Rm[4], Rm5 = Rm[5];
        const float Rm6 = Rm[6], Rm7 = Rm[7], Rm8 = Rm[8];
        const int gm = colBase + mloc;

#pragma unroll
        for (int r = 0; r < 8; ++r) {
            const int nloc = sn * 16 + 8 * h + r;   // matches D layout
            const int gn   = rowBase + nloc;
            float dist2 = fmaxf(d2[r], 0.f) + EPSF; // clamp Gram cancellation
            float invd  = __builtin_amdgcn_rsqf(dist2);
            float dist  = dist2 * invd;             // sqrt(dist2)
            const float4 pn  = posr[nloc];          // broadcast per half-wave
            const float4 snv = scr[nloc];
            float dx = (pn.x - pm.x) * invd;        // normalized direction
            float dy = (pn.y - pm.y) * invd;
            float dz = (pn.z - pm.z) * invd;
            // r_dir[n][m]: rotate by R[n], scale by scales[m]
            const float* Rn = &Rr[nloc * 12];       // uniform per half-wave
            float u0 = dx * Rn[0] + dy * Rn[3] + dz * Rn[6];
            float u1 = dx * Rn[1] + dy * Rn[4] + dz * Rn[7];
            float u2 = dx * Rn[2] + dy * Rn[5] + dz * Rn[8];
            u0 *= smv.x; u1 *= smv.y; u2 *= smv.z;
            float s1 = u0 * u0 + u1 * u1 + u2 * u2;
            // r_dir[m][n]: direction negates but is squared -> reuse d
            float v0 = dx * Rm0 + dy * Rm3 + dz * Rm6;
            float v1 = dx * Rm1 + dy * Rm4 + dz * Rm7;
            float v2 = dx * Rm2 + dy * Rm5 + dz * Rm8;
            v0 *= snv.x; v1 *= snv.y; v2 *= snv.z;
            float s2 = v0 * v0 + v1 * v1 + v2 * v2;
            float rt = __builtin_amdgcn_sqrtf(s1) + __builtin_amdgcn_sqrtf(s2);
            float ov = fmaxf(rt - dist, 0.f);
            float rep = (ov * ov) * __builtin_amdgcn_rcpf(fmaf(0.1f, ov, 1.f));
            if (gn == gm) rep = 0.f;                // diagonal mask
            acc += rep;
        }
    }

    // deterministic block reduction
    red[tid] = acc;
    __syncthreads();
    for (int s = 128; s > 0; s >>= 1) {
        if (tid < s) red[tid] += red[tid + s];
        __syncthreads();
    }
    if (tid == 0) {
        int bid = (blockIdx.z * gridDim.y + blockIdx.y) * gridDim.x + blockIdx.x;
        partials[bid] = red[0];
    }
}

// ---------------------------------------------------------------------------
// Kernel 3: deterministic final reduction + mean scaling.
// ---------------------------------------------------------------------------
__global__ void __launch_bounds__(256)
reduce_kernel(const float* __restrict__ partials, int n, float scale,
              float* __restrict__ out) {
    __shared__ float red[256];
    float s = 0.f;
    for (int i = threadIdx.x; i < n; i += 256) s += partials[i];
    red[threadIdx.x] = s;
    __syncthreads();
    for (int k = 128; k > 0; k >>= 1) {
        if (threadIdx.x < k) red[threadIdx.x] += red[threadIdx.x + k];
        __syncthreads();
    }
    if (threadIdx.x == 0) out[0] = red[0] * scale;
}

// ---------------------------------------------------------------------------
extern "C" void kernel_launch(void* const* d_in, const int* in_sizes, int n_in,
                              void* d_out, int out_size, void* d_ws,
                              size_t ws_size, hipStream_t stream) {
    const float* xyz       = (const float*)d_in[0];
    const float* scales    = (const float*)d_in[1];
    const float* rotations = (const float*)d_in[2];

    const int B = 2;                       // per reference setup
    const int N = in_sizes[0] / (3 * B);   // 4096
    const int P = B * N;

    float* Rg       = (float*)d_ws;                 // P * 12 floats
    float* partials = Rg + (size_t)P * 12;          // one per block

    quat2rot_kernel<<<(P + 255) / 256, 256, 0, stream>>>(rotations, Rg, P);

    dim3 grid(N / TILE, N / TILE, B);
    tile_kernel<<<grid, 256, 0, stream>>>(xyz, scales, Rg, partials, N);

    const int nparts = (N / TILE) * (N / TILE) * B;
    const float scale = 1.0f / ((float)B * (float)N * (float)N); // 2^-25 exact
    reduce_kernel<<<1, 256, 0, stream>>>(partials, nparts, scale, (float*)d_out);
}